// PredictYOLO_v2_20358144983135
// MI455X (gfx1250) — compile-verified
//
#include <hip/hip_runtime.h>
#include <math.h>
#include <stdint.h>

#define BATCH   32
#define GW      52
#define GH      52
#define HW      (GW*GH)           // 2704
#define NANC    5
#define NCLS    20
#define DIM     (1+NCLS+4)        // 25
#define NTOT    (BATCH*HW*NANC)   // 432640  (divisible by 4)
#define NV4     (NTOT/4)          // 108160 vec4 groups
#define CONF_THR 0.5f
#define NMS_THR  0.5f
#define MAXDET  200
#define TPB     256
#define NBLK    256               // must equal TPB for the reduce kernel

__constant__ float c_aw[NANC] = {0.08f, 0.26f, 0.51f, 0.72f, 0.98f};
__constant__ float c_ah[NANC] = {0.09f, 0.34f, 0.88f, 0.39f, 0.81f};

__device__ __forceinline__ float sigmoidf_fast(float x) {
    return 1.0f / (1.0f + __expf(-x));
}

// ---------------------------------------------------------------------------
// Init: chosen-box slot (idx = -1 disables suppression on NMS step 0)
// ---------------------------------------------------------------------------
__global__ void init_kernel(float* chosen) {
    if (threadIdx.x == 0) {
        ((int*)chosen)[0] = -1;
        chosen[1] = 0.f; chosen[2] = 0.f; chosen[3] = 0.f; chosen[4] = 0.f;
    }
}

// ---------------------------------------------------------------------------
// Decode: thread t = (b*NANC + a)*HW + hw  -> fully coalesced per-channel loads
// pyolos element (b, c=d*5+a, hw) at  (b*125 + d*5 + a)*HW + hw
// ---------------------------------------------------------------------------
__global__ void decode_kernel(const float* __restrict__ p,
                              float* __restrict__ s,
                              float* __restrict__ boxo,   // offset boxes (NMS)
                              float* __restrict__ boxs,   // original boxes (output)
                              float* __restrict__ lbl)
{
    int t = blockIdx.x * blockDim.x + threadIdx.x;
    if (t >= NTOT) return;
    int hw = t % HW;
    int ba = t / HW;
    int a  = ba % NANC;
    int b  = ba / NANC;

    const float* base = p + (size_t)(b*DIM*NANC + a) * HW + hw;
    const int CST = NANC * HW;   // channel stride in elements

    // prefetch the box-regression channels while we chew the class channels
    __builtin_prefetch(base + (size_t)(1+NCLS)*CST, 0, 0);

    float v0 = base[0];
    float rawmax = base[CST];
    int   am = 1;                 // label is argmax+1, classes start at d=1
    #pragma unroll
    for (int d = 2; d <= NCLS; ++d) {
        float v = base[(size_t)d * CST];
        if (v > rawmax) { rawmax = v; am = d; }   // strict > : first max wins
    }
    float t0 = base[(size_t)(1+NCLS+0) * CST];
    float t1 = base[(size_t)(1+NCLS+1) * CST];
    float t2 = base[(size_t)(1+NCLS+2) * CST];
    float t3 = base[(size_t)(1+NCLS+3) * CST];

    float pconf = sigmoidf_fast(v0);
    float cmax  = sigmoidf_fast(rawmax);     // sigmoid monotonic: one sigmoid suffices
    float score = pconf * cmax;
    float sc    = (score > CONF_THR) ? score : -1.0f;

    float gx = (float)(hw % GW), gy = (float)(hw / GW);
    float x  = (sigmoidf_fast(t0) + gx) * (1.0f / GW);
    float y  = (sigmoidf_fast(t1) + gy) * (1.0f / GH);
    float w  = __expf(t2) * c_aw[a];
    float h  = __expf(t3) * c_ah[a];

    float l  = fminf(fmaxf(x - 0.5f*w, 0.f), 1.f);
    float tt = fminf(fmaxf(y - 0.5f*h, 0.f), 1.f);
    float r  = fminf(fmaxf(x + 0.5f*w, 0.f), 1.f);
    float bb = fminf(fmaxf(y + 0.5f*h, 0.f), 1.f);

    float off = (float)(b*(NCLS+1) + am) * 2.0f;   // class+batch aware NMS offset
    int n = (b*HW + hw)*NANC + a;                  // reference flatten order

    s[n] = sc;
    boxs[4*n+0] = l;        boxs[4*n+1] = tt;
    boxs[4*n+2] = r;        boxs[4*n+3] = bb;
    boxo[4*n+0] = l  + off; boxo[4*n+1] = tt + off;
    boxo[4*n+2] = r  + off; boxo[4*n+3] = bb + off;
    lbl[n] = (float)am;
}

// ---------------------------------------------------------------------------
// NMS pass: apply suppression from previous pick, block-level argmax.
// Scores stream global -> LDS via 16B/lane async copies (ASYNCcnt-tracked,
// double buffered with s_wait_asynccnt). wave32 shfl reductions.
// Each thread owns 4 consecutive scores per iteration; NTOT % 4 == 0.
// ---------------------------------------------------------------------------
__global__ void nms_pass_kernel(float* __restrict__ s,
                                const float* __restrict__ boxo,
                                const float* __restrict__ chosen,
                                float* __restrict__ pmax,
                                int*   __restrict__ pidx)
{
    __shared__ __align__(16) float tile[2][TPB][4];
    __shared__ float wmaxs[TPB/32];
    __shared__ int   widxs[TPB/32];

    const int tid  = threadIdx.x;
    const int cidx = ((const int*)chosen)[0];
    const float cl = chosen[1], ct = chosen[2], cr = chosen[3], cb = chosen[4];
    const float carea = fmaxf(cr - cl, 0.f) * fmaxf(cb - ct, 0.f);
    const uint64_t sbase = (uint64_t)(uintptr_t)s;

    float best = -INFINITY;
    int   bidx = 0x7fffffff;

    const int idx0   = blockIdx.x * TPB + tid;           // vec4-group index
    const int stride = NBLK * TPB;                       // 65536 vec4 groups
    const int nit    = (NV4 + stride - 1) / stride;      // 2

    // prime tile 0: async 16B/lane copy global -> LDS (tracked by ASYNCcnt)
    {
        int ld = (idx0 < NV4) ? idx0 : (NV4 - 1);
        uint32_t lds = (uint32_t)(uintptr_t)&tile[0][tid][0];
        asm volatile("global_load_async_to_lds_b128 %0, %1, %2"
                     :: "v"(lds), "v"(ld * 16), "s"(sbase) : "memory");
    }

    for (int it = 0; it < nit; ++it) {
        int buf = it & 1;
        if (it + 1 < nit) {
            int nidx = idx0 + (it + 1) * stride;
            int ld   = (nidx < NV4) ? nidx : (NV4 - 1);
            uint32_t lds = (uint32_t)(uintptr_t)&tile[buf ^ 1][tid][0];
            asm volatile("global_load_async_to_lds_b128 %0, %1, %2"
                         :: "v"(lds), "v"(ld * 16), "s"(sbase) : "memory");
            asm volatile("s_wait_asynccnt 0x1" ::: "memory");  // tile `it` done
        } else {
            asm volatile("s_wait_asynccnt 0x0" ::: "memory");
        }

        int nv = idx0 + it * stride;
        if (nv < NV4) {
            #pragma unroll
            for (int e = 0; e < 4; ++e) {
                int   j = nv * 4 + e;
                float v = tile[buf][tid][e];
                if (j == cidx) {
                    v = -INFINITY; s[j] = v;
                } else if (cidx >= 0 && v > 0.0f) {
                    // only positive-score boxes can affect the kept set
                    const float* bo = boxo + 4*(size_t)j;
                    float b0 = bo[0], b1 = bo[1], b2 = bo[2], b3 = bo[3];
                    float iw = fmaxf(fminf(cr, b2) - fmaxf(cl, b0), 0.f);
                    float ih = fmaxf(fminf(cb, b3) - fmaxf(ct, b1), 0.f);
                    float inter = iw * ih;
                    float a2 = fmaxf(b2 - b0, 0.f) * fmaxf(b3 - b1, 0.f);
                    float iou = inter / (carea + a2 - inter + 1e-9f);
                    if (iou > NMS_THR) { v = -INFINITY; s[j] = v; }
                }
                if (v > best || (v == best && j < bidx)) { best = v; bidx = j; }
            }
        }
    }

    // wave32 reduction
    #pragma unroll
    for (int m = 16; m; m >>= 1) {
        float ov = __shfl_xor(best, m, 32);
        int   oi = __shfl_xor(bidx, m, 32);
        if (ov > best || (ov == best && oi < bidx)) { best = ov; bidx = oi; }
    }
    int wid = tid >> 5;
    if ((tid & 31) == 0) { wmaxs[wid] = best; widxs[wid] = bidx; }
    __syncthreads();
    if (wid == 0) {
        if (tid < TPB/32) { best = wmaxs[tid]; bidx = widxs[tid]; }
        else              { best = -INFINITY;  bidx = 0x7fffffff; }
        #pragma unroll
        for (int m = 16; m; m >>= 1) {
            float ov = __shfl_xor(best, m, 32);
            int   oi = __shfl_xor(bidx, m, 32);
            if (ov > best || (ov == best && oi < bidx)) { best = ov; bidx = oi; }
        }
        if (tid == 0) { pmax[blockIdx.x] = best; pidx[blockIdx.x] = bidx; }
    }
}

// ---------------------------------------------------------------------------
// Reduce NBLK partials -> global argmax; record pick; publish chosen box.
// ---------------------------------------------------------------------------
__global__ void nms_reduce_kernel(const float* __restrict__ pmax,
                                  const int*   __restrict__ pidx,
                                  const float* __restrict__ boxo,
                                  float* __restrict__ chosen,
                                  float* __restrict__ selVal,
                                  int*   __restrict__ selIdx,
                                  int step)
{
    __shared__ float wmaxs[TPB/32];
    __shared__ int   widxs[TPB/32];
    int tid = threadIdx.x;              // TPB == NBLK
    float best = pmax[tid];
    int   bidx = pidx[tid];
    #pragma unroll
    for (int m = 16; m; m >>= 1) {
        float ov = __shfl_xor(best, m, 32);
        int   oi = __shfl_xor(bidx, m, 32);
        if (ov > best || (ov == best && oi < bidx)) { best = ov; bidx = oi; }
    }
    int wid = tid >> 5;
    if ((tid & 31) == 0) { wmaxs[wid] = best; widxs[wid] = bidx; }
    __syncthreads();
    if (wid == 0) {
        if (tid < TPB/32) { best = wmaxs[tid]; bidx = widxs[tid]; }
        else              { best = -INFINITY;  bidx = 0x7fffffff; }
        #pragma unroll
        for (int m = 16; m; m >>= 1) {
            float ov = __shfl_xor(best, m, 32);
            int   oi = __shfl_xor(bidx, m, 32);
            if (ov > best || (ov == best && oi < bidx)) { best = ov; bidx = oi; }
        }
        if (tid == 0) {
            selVal[step] = best;
            selIdx[step] = bidx;
            ((int*)chosen)[0] = bidx;
            chosen[1] = boxo[4*(size_t)bidx+0];
            chosen[2] = boxo[4*(size_t)bidx+1];
            chosen[3] = boxo[4*(size_t)bidx+2];
            chosen[4] = boxo[4*(size_t)bidx+3];
        }
    }
}

// ---------------------------------------------------------------------------
// Finalize: emit (ids_batch2[200], boxes[200*4], labels[200], scores[200],
//                 keep[200]) all as float, in reference return order.
// ---------------------------------------------------------------------------
__global__ void finalize_kernel(const float* __restrict__ selVal,
                                const int*   __restrict__ selIdx,
                                const float* __restrict__ boxs,
                                const float* __restrict__ lbl,
                                float* __restrict__ out)
{
    int k = blockIdx.x * blockDim.x + threadIdx.x;
    if (k >= MAXDET) return;
    float sv  = selVal[k];
    int   i   = selIdx[k];
    bool keep = sv > 0.0f;
    int   b   = i / (HW * NANC);
    float lb  = lbl[i];
    out[k]                 = keep ? (float)b : -1.0f;
    out[MAXDET + 4*k + 0]  = keep ? boxs[4*(size_t)i+0] : 0.0f;
    out[MAXDET + 4*k + 1]  = keep ? boxs[4*(size_t)i+1] : 0.0f;
    out[MAXDET + 4*k + 2]  = keep ? boxs[4*(size_t)i+2] : 0.0f;
    out[MAXDET + 4*k + 3]  = keep ? boxs[4*(size_t)i+3] : 0.0f;
    out[MAXDET*5 + k]      = keep ? lb  : 0.0f;
    out[MAXDET*6 + k]      = keep ? sv  : 0.0f;
    out[MAXDET*7 + k]      = keep ? 1.0f : 0.0f;
}

// ---------------------------------------------------------------------------
extern "C" void kernel_launch(void* const* d_in, const int* in_sizes, int n_in,
                              void* d_out, int out_size, void* d_ws, size_t ws_size,
                              hipStream_t stream)
{
    (void)in_sizes; (void)n_in; (void)out_size; (void)ws_size;
    const float* pyolos = (const float*)d_in[0];
    float* out = (float*)d_out;

    float* ws     = (float*)d_ws;
    float* s      = ws;                         // [N]
    float* boxo   = ws + (size_t)NTOT;          // [4N]  offset boxes (NMS)
    float* boxs   = ws + (size_t)5*NTOT;        // [4N]  original boxes
    float* lbl    = ws + (size_t)9*NTOT;        // [N]
    float* pmax   = ws + (size_t)10*NTOT;       // [NBLK]
    int*   pidx   = (int*)(ws + (size_t)10*NTOT + NBLK);     // [NBLK]
    float* selVal = ws + (size_t)10*NTOT + 2*NBLK;           // [MAXDET]
    int*   selIdx = (int*)(ws + (size_t)10*NTOT + 2*NBLK + MAXDET); // [MAXDET]
    float* chosen = ws + (size_t)10*NTOT + 2*NBLK + 2*MAXDET;       // [5]

    init_kernel<<<1, 32, 0, stream>>>(chosen);
    decode_kernel<<<(NTOT + TPB - 1)/TPB, TPB, 0, stream>>>(pyolos, s, boxo, boxs, lbl);

    for (int k = 0; k < MAXDET; ++k) {
        nms_pass_kernel<<<NBLK, TPB, 0, stream>>>(s, boxo, chosen, pmax, pidx);
        nms_reduce_kernel<<<1, TPB, 0, stream>>>(pmax, pidx, boxo, chosen,
                                                 selVal, selIdx, k);
    }
    finalize_kernel<<<1, TPB, 0, stream>>>(selVal, selIdx, boxs, lbl, out);
}